// LigandGNN_45595372815095
// MI455X (gfx1250) — compile-verified
//
#include <hip/hip_runtime.h>

#define N_NODES   50000
#define N_EDGES   600000
#define N_GRAPHS  1000
#define IN_DIM    16
#define HID       128
#define DESC      17
#define NUM_LAYERS 3
#define BN_EPS    1e-5f

typedef float v2f __attribute__((ext_vector_type(2)));
typedef float v8f __attribute__((ext_vector_type(8)));

// ---------------------------------------------------------------- utilities
__global__ void zero_kernel(float* __restrict__ p, int n) {
  int i = blockIdx.x * blockDim.x + threadIdx.x;
  if (i < n) p[i] = 0.f;
}

// ------------------------------------------------- h = x @ Wp + bp  (K=16)
__global__ void proj_kernel(const float* __restrict__ x,
                            const float* __restrict__ Wp,
                            const float* __restrict__ bp,
                            float* __restrict__ h) {
  int i = blockIdx.x * blockDim.x + threadIdx.x; // node*128 + c
  if (i >= N_NODES * HID) return;
  int n = i >> 7, c = i & 127;
  const float* xr = x + (size_t)n * IN_DIM;
  float acc = bp[c];
#pragma unroll
  for (int k = 0; k < IN_DIM; ++k) acc = fmaf(xr[k], Wp[k * HID + c], acc);
  h[i] = acc;
}

// ------------------- aggr[dst] += relu(h[src] + edge_attr)  (1 wave / edge)
__global__ void edge_kernel(const float* __restrict__ h,
                            const int* __restrict__ ei,
                            const float* __restrict__ ea,
                            float* __restrict__ aggr) {
  int t = blockIdx.x * blockDim.x + threadIdx.x;
  int e = t >> 5;
  if (e >= N_EDGES) return;
  int q = (t & 31) << 2;                 // 4 channels per lane
  int src = ei[e];
  int dst = ei[N_EDGES + e];
  float4 hv = *(const float4*)(h + (size_t)src * HID + q);
  float4 ev = *(const float4*)(ea + (size_t)e * HID + q);
  float m0 = fmaxf(hv.x + ev.x, 0.f);
  float m1 = fmaxf(hv.y + ev.y, 0.f);
  float m2 = fmaxf(hv.z + ev.z, 0.f);
  float m3 = fmaxf(hv.w + ev.w, 0.f);
  float* d = aggr + (size_t)dst * HID + q;
  atomicAdd(d + 0, m0);
  atomicAdd(d + 1, m1);
  atomicAdd(d + 2, m2);
  atomicAdd(d + 3, m3);
}

// -------- fused: z = h+aggr; t = relu(z@W1+b1); y = t@W2+b2; BN statistics
// WG = 256 threads = 8 waves; WG owns 64 node rows; wave w owns N-cols
// [16w,16w+16) for 4 M-subtiles of 16 rows each. fp32 WMMA 16x16x4, K=128.
#define MROWS 64
#define LSTR  132   // 128 + 4 pad floats -> no LDS bank conflicts

__global__ __launch_bounds__(256) void mlp_kernel(
    const float* __restrict__ h, const float* __restrict__ aggr,
    const float* __restrict__ W1, const float* __restrict__ b1,
    const float* __restrict__ W2, const float* __restrict__ b2,
    float* __restrict__ y, float* __restrict__ stats) {
  __shared__ float panel[MROWS * LSTR];  // 33.8 KB

  const int row0 = blockIdx.x * MROWS;
  const int tid  = threadIdx.x;
  const int wave = tid >> 5;
  const int lane = tid & 31;
  const int lm   = lane & 15;   // N index within tile / M row for A-frag
  const int lh   = lane >> 4;   // half-wave selector
  const int n0   = wave << 4;

  // ---- stage z = h + aggr into LDS (64 x 128, padded) ----
  for (int i = tid; i < MROWS * (HID / 4); i += 256) {
    int r = i >> 5;             // 32 float4 per row
    int c = (i & 31) << 2;
    int row = row0 + r;
    float zx = 0.f, zy = 0.f, zz = 0.f, zw = 0.f;
    if (row < N_NODES) {
      float4 a = *(const float4*)(h    + (size_t)row * HID + c);
      float4 b = *(const float4*)(aggr + (size_t)row * HID + c);
      zx = a.x + b.x; zy = a.y + b.y; zz = a.z + b.z; zw = a.w + b.w;
    }
    float* dp = &panel[r * LSTR + c];
    dp[0] = zx; dp[1] = zy; dp[2] = zz; dp[3] = zw;
  }
  __syncthreads();

  // ---- GEMM1: t = relu(z @ W1 + b1) ----
  v8f acc[4];
#pragma unroll
  for (int mt = 0; mt < 4; ++mt) acc[mt] = (v8f){0,0,0,0,0,0,0,0};

  for (int kb = 0; kb < HID / 4; ++kb) {
    const int kA = (kb << 2) + (lh << 1);       // A/B K columns for this lane
    v2f bfrag;
    bfrag.x = W1[(size_t)kA * HID + n0 + lm];
    bfrag.y = W1[(size_t)(kA + 1) * HID + n0 + lm];
#pragma unroll
    for (int mt = 0; mt < 4; ++mt) {
      v2f afrag = *(const v2f*)&panel[(mt * 16 + lm) * LSTR + kA];
      acc[mt] = __builtin_amdgcn_wmma_f32_16x16x4_f32(
          false, afrag, false, bfrag, (short)0, acc[mt], false, false);
    }
  }
  const float bias1 = b1[n0 + lm];
  __syncthreads();   // all z reads done -> panel reusable for t

  // relu + bias, write t back into panel (C layout: M = r + 8*lh, N = lm)
#pragma unroll
  for (int mt = 0; mt < 4; ++mt)
#pragma unroll
    for (int r = 0; r < 8; ++r)
      panel[(mt * 16 + r + (lh << 3)) * LSTR + n0 + lm] =
          fmaxf(acc[mt][r] + bias1, 0.f);
  __syncthreads();

  // ---- GEMM2: y = t @ W2 + b2 ----
#pragma unroll
  for (int mt = 0; mt < 4; ++mt) acc[mt] = (v8f){0,0,0,0,0,0,0,0};

  for (int kb = 0; kb < HID / 4; ++kb) {
    const int kA = (kb << 2) + (lh << 1);
    v2f bfrag;
    bfrag.x = W2[(size_t)kA * HID + n0 + lm];
    bfrag.y = W2[(size_t)(kA + 1) * HID + n0 + lm];
#pragma unroll
    for (int mt = 0; mt < 4; ++mt) {
      v2f afrag = *(const v2f*)&panel[(mt * 16 + lm) * LSTR + kA];
      acc[mt] = __builtin_amdgcn_wmma_f32_16x16x4_f32(
          false, afrag, false, bfrag, (short)0, acc[mt], false, false);
    }
  }
  const float bias2 = b2[n0 + lm];

  // store y + per-channel BN partial sums (this lane covers 32 rows, 1 chan)
  float psum = 0.f, psq = 0.f;
#pragma unroll
  for (int mt = 0; mt < 4; ++mt)
#pragma unroll
    for (int r = 0; r < 8; ++r) {
      int row = row0 + mt * 16 + r + (lh << 3);
      float v = acc[mt][r] + bias2;
      if (row < N_NODES) {
        y[(size_t)row * HID + n0 + lm] = v;
        psum += v;
        psq  += v * v;
      }
    }
  atomicAdd(&stats[n0 + lm], psum);
  atomicAdd(&stats[HID + n0 + lm], psq);
}

// -------------------------------------------- batch-norm (train stats)+relu
__global__ void bn_relu_kernel(const float* __restrict__ y,
                               const float* __restrict__ stats,
                               const float* __restrict__ gamma,
                               const float* __restrict__ beta,
                               float* __restrict__ h) {
  int i = blockIdx.x * blockDim.x + threadIdx.x;
  if (i >= N_NODES * HID) return;
  int c = i & 127;
  const float invN = 1.f / (float)N_NODES;
  float mean = stats[c] * invN;
  float var  = stats[HID + c] * invN - mean * mean;
  float inv  = rsqrtf(var + BN_EPS);
  float v = (y[i] - mean) * inv * gamma[c] + beta[c];
  h[i] = fmaxf(v, 0.f);
}

// ------------------------------------------------------- global mean pool
__global__ void pool_kernel(const float* __restrict__ h,
                            const int* __restrict__ batch,
                            float* __restrict__ pooled,
                            float* __restrict__ counts) {
  int t = blockIdx.x * blockDim.x + threadIdx.x;
  int n = t >> 5;
  if (n >= N_NODES) return;
  int q = (t & 31) << 2;
  int g = batch[n];
  float4 hv = *(const float4*)(h + (size_t)n * HID + q);
  float* d = pooled + (size_t)g * HID + q;
  atomicAdd(d + 0, hv.x);
  atomicAdd(d + 1, hv.y);
  atomicAdd(d + 2, hv.z);
  atomicAdd(d + 3, hv.w);
  if ((t & 31) == 0) atomicAdd(&counts[g], 1.f);
}

__global__ void final_kernel(const float* __restrict__ pooled,
                             const float* __restrict__ counts,
                             const float* __restrict__ mol,
                             const float* __restrict__ Wf,
                             const float* __restrict__ bfp,
                             float* __restrict__ out) {
  int g = blockIdx.x * blockDim.x + threadIdx.x;
  if (g >= N_GRAPHS) return;
  float inv = 1.f / fmaxf(counts[g], 1.f);
  float acc = bfp[0];
  for (int c = 0; c < HID; ++c)
    acc = fmaf(pooled[(size_t)g * HID + c] * inv, Wf[c], acc);
  for (int d = 0; d < DESC; ++d)
    acc = fmaf(mol[(size_t)g * DESC + d], Wf[HID + d], acc);
  out[g] = acc;
}

// --------------------------------------------------------------------------
extern "C" void kernel_launch(void* const* d_in, const int* in_sizes, int n_in,
                              void* d_out, int out_size, void* d_ws, size_t ws_size,
                              hipStream_t stream) {
  const float* x     = (const float*)d_in[0];
  const int*   ei    = (const int*)  d_in[1];
  const float* ea    = (const float*)d_in[2];
  const int*   batch = (const int*)  d_in[3];
  const float* mol   = (const float*)d_in[4];
  const float* Wp    = (const float*)d_in[5];
  const float* bp    = (const float*)d_in[6];
  const float* W1    = (const float*)d_in[7];
  const float* b1    = (const float*)d_in[8];
  const float* W2    = (const float*)d_in[9];
  const float* b2    = (const float*)d_in[10];
  const float* gamma = (const float*)d_in[11];
  const float* beta  = (const float*)d_in[12];
  const float* Wf    = (const float*)d_in[13];
  const float* bf    = (const float*)d_in[14];
  float* out = (float*)d_out;

  // workspace layout (floats): h | aggr | y | stats(256) | pooled | counts
  const size_t NH = (size_t)N_NODES * HID;   // 6.4M
  float* ws     = (float*)d_ws;
  float* h      = ws;
  float* aggr   = ws + NH;
  float* ybuf   = ws + 2 * NH;
  float* stats  = ws + 3 * NH;
  float* pooled = stats + 2 * HID;
  float* counts = pooled + (size_t)N_GRAPHS * HID;

  dim3 b256(256);
  proj_kernel<<<(N_NODES * HID + 255) / 256, b256, 0, stream>>>(x, Wp, bp, h);

  for (int l = 0; l < NUM_LAYERS; ++l) {
    zero_kernel<<<(int)((NH + 255) / 256), b256, 0, stream>>>(aggr, (int)NH);
    zero_kernel<<<1, b256, 0, stream>>>(stats, 2 * HID);
    edge_kernel<<<(N_EDGES * 32 + 255) / 256, b256, 0, stream>>>(h, ei, ea, aggr);
    mlp_kernel<<<(N_NODES + MROWS - 1) / MROWS, b256, 0, stream>>>(
        h, aggr,
        W1 + (size_t)l * HID * HID, b1 + (size_t)l * HID,
        W2 + (size_t)l * HID * HID, b2 + (size_t)l * HID,
        ybuf, stats);
    bn_relu_kernel<<<(N_NODES * HID + 255) / 256, b256, 0, stream>>>(
        ybuf, stats, gamma + (size_t)l * HID, beta + (size_t)l * HID, h);
  }

  zero_kernel<<<(N_GRAPHS * HID + N_GRAPHS + 255) / 256, b256, 0, stream>>>(
      pooled, N_GRAPHS * HID + N_GRAPHS);
  pool_kernel<<<(N_NODES * 32 + 255) / 256, b256, 0, stream>>>(h, batch, pooled, counts);
  final_kernel<<<(N_GRAPHS + 255) / 256, b256, 0, stream>>>(pooled, counts, mol, Wf, bf, out);
}